// HybridModel_23828478558270
// MI455X (gfx1250) — compile-verified
//
#include <hip/hip_runtime.h>
#include <math.h>

// ---------------------------------------------------------------------------
// Problem constants (from setup_inputs):
//   B=8192, D_IN=256, HID=128 (only first 10 used), NQ=10, NL=6, D_OUT=64
// ---------------------------------------------------------------------------
#define NSAMP   8192
#define DIN     256
#define NQ      10
#define NL      6
#define DOUT    64
#define FEAT_LD 16   // we compute 16 hidden units (>=10 needed), padded stride

typedef __attribute__((ext_vector_type(2))) float v2f;
typedef __attribute__((ext_vector_type(8))) float v8f;

// ===========================================================================
// Kernel 1: feats = relu(x @ W1[0:16].T + b1[0:16])  via V_WMMA_F32_16X16X4_F32
//   One wave computes a 16(sample) x 16(hidden) f32 tile, K stepped by 4.
//   A (16x4, f32, 2 VGPR): lanes 0-15 -> M=lane, K=k+0/k+1 ; lanes 16-31 -> K=k+2/k+3
//   B (4x16, f32, 2 VGPR): lanes 0-15 -> N=lane, K=k+0/k+1 ; lanes 16-31 -> K=k+2/k+3
//   D (16x16, 8 VGPR): vgpr r -> M = r + (lane>=16 ? 8 : 0), N = lane & 15
// ===========================================================================
__global__ __launch_bounds__(256) void gemm1_relu_wmma(
    const float* __restrict__ x,   // (8192, 256)
    const float* __restrict__ W1,  // (128, 256) row-major; rows 0..15 used
    const float* __restrict__ b1,  // (128)
    float* __restrict__ feats)     // (8192, 16) scratch
{
  const int lane = threadIdx.x & 31;
  const int wave = threadIdx.x >> 5;
  const int tile = blockIdx.x * 8 + wave;      // 512 tiles of 16 samples
  const int m0   = tile * 16;
  const int half = lane >> 4;                  // 0: K pair {0,1}, 1: K pair {2,3}
  const int lr   = lane & 15;

  const float* xrow = x  + (size_t)(m0 + lr) * DIN + half * 2;
  const float* wrow = W1 + (size_t)lr        * DIN + half * 2;

  v8f acc = {};
  for (int k = 0; k < DIN; k += 4) {
    v2f a, b;
    a[0] = xrow[k];  a[1] = xrow[k + 1];
    b[0] = wrow[k];  b[1] = wrow[k + 1];
    // 8 args: (neg_a, A, neg_b, B, c_mod, C, reuse_a, reuse_b)
    acc = __builtin_amdgcn_wmma_f32_16x16x4_f32(false, a, false, b,
                                                (short)0, acc, false, false);
  }

  const float bias = b1[lr];
  #pragma unroll
  for (int r = 0; r < 8; ++r) {
    const int m = r + half * 8;
    feats[(size_t)(m0 + m) * FEAT_LD + lr] = fmaxf(acc[r] + bias, 0.0f);
  }
}

// ===========================================================================
// Kernel 2: 10-qubit statevector sim, one wave32 per sample.
//   Flat index i (10 bits), qubit q <-> bit p = 9-q (wire 0 = MSB).
//   lane = i >> 5 (bits 9..5, qubits 0..4), reg = i & 31 (bits 4..0, qubits 5..9).
//   State: float2 st[32] per lane = 64 VGPRs; all reg indices compile-time.
// ===========================================================================
__device__ __forceinline__ float2 shfl_xor_c(float2 v, int m) {
  float2 r;
  r.x = __shfl_xor(v.x, m, 32);
  r.y = __shfl_xor(v.y, m, 32);
  return r;
}

// Apply 2x2 complex gate U=[[u00,u01],[u10,u11]] on flat bit P.
template <int P>
__device__ __forceinline__ void apply_1q(float2 (&st)[32], int lane,
                                         float2 u00, float2 u01,
                                         float2 u10, float2 u11) {
  if constexpr (P >= 5) {                      // lane bit: cross-lane butterfly
    constexpr int m = 1 << (P - 5);
    const bool hi = (lane & m) != 0;
    const float2 cm = hi ? u11 : u00;          // coeff of own amplitude
    const float2 co = hi ? u10 : u01;          // coeff of partner amplitude
    #pragma unroll
    for (int r = 0; r < 32; ++r) {
      float2 o = shfl_xor_c(st[r], m);
      float2 s = st[r], n;
      n.x = cm.x * s.x - cm.y * s.y + co.x * o.x - co.y * o.y;
      n.y = cm.x * s.y + cm.y * s.x + co.x * o.y + co.y * o.x;
      st[r] = n;
    }
  } else {                                     // register bit: in-register pairs
    constexpr int m = 1 << P;
    #pragma unroll
    for (int r = 0; r < 32; ++r) {
      if ((r & m) == 0) {
        float2 lo = st[r], hh = st[r + m];
        float2 a, b;
        a.x = u00.x * lo.x - u00.y * lo.y + u01.x * hh.x - u01.y * hh.y;
        a.y = u00.x * lo.y + u00.y * lo.x + u01.x * hh.y + u01.y * hh.x;
        b.x = u10.x * lo.x - u10.y * lo.y + u11.x * hh.x - u11.y * hh.y;
        b.y = u10.x * lo.y + u10.y * lo.x + u11.x * hh.y + u11.y * hh.x;
        st[r] = a; st[r + m] = b;
      }
    }
  }
}

// CNOT: swap along target bit PT where control bit PC == 1.
template <int PC, int PT>
__device__ __forceinline__ void apply_cnot(float2 (&st)[32], int lane) {
  if constexpr (PT >= 5) {                     // target is a lane bit
    constexpr int tm = 1 << (PT - 5);
    if constexpr (PC >= 5) {
      const bool ctrl = (lane & (1 << (PC - 5))) != 0;
      #pragma unroll
      for (int r = 0; r < 32; ++r) {
        float2 o = shfl_xor_c(st[r], tm);      // all lanes shuffle
        if (ctrl) st[r] = o;                   // EXEC-masked select
      }
    } else {
      #pragma unroll
      for (int r = 0; r < 32; ++r) {
        float2 o = shfl_xor_c(st[r], tm);
        if (r & (1 << PC)) st[r] = o;          // compile-time per-reg condition
      }
    }
  } else {                                     // target is a register bit
    constexpr int tm = 1 << PT;
    if constexpr (PC >= 5) {
      const bool ctrl = (lane & (1 << (PC - 5))) != 0;
      #pragma unroll
      for (int r = 0; r < 32; ++r) {
        if ((r & tm) == 0) {
          float2 a = st[r], c = st[r + tm];
          st[r]      = ctrl ? c : a;
          st[r + tm] = ctrl ? a : c;
        }
      }
    } else {
      constexpr int cm = 1 << PC;
      #pragma unroll
      for (int r = 0; r < 32; ++r) {
        if ((r & cm) && !(r & tm)) {           // pure register renaming
          float2 t = st[r]; st[r] = st[r + tm]; st[r + tm] = t;
        }
      }
    }
  }
}

__device__ __forceinline__ void rot_mat(float phi, float th, float om,
                                        float2& m00, float2& m01,
                                        float2& m10, float2& m11) {
  float c, s, ca, sa, cb, sb;
  __sincosf(th * 0.5f, &s, &c);
  __sincosf(0.5f * (phi + om), &sa, &ca);
  __sincosf(0.5f * (phi - om), &sb, &cb);
  m00 = make_float2( ca * c, -sa * c);   // e^{-i(phi+om)/2} cos
  m01 = make_float2(-cb * s, -sb * s);   // -e^{+i(phi-om)/2} sin
  m10 = make_float2( cb * s, -sb * s);   // e^{-i(phi-om)/2} sin
  m11 = make_float2( ca * c,  sa * c);   // e^{+i(phi+om)/2} cos
}

template <int Q>
__device__ __forceinline__ void embed_rx(float2 (&st)[32],
                                         const float* __restrict__ f, int lane) {
  if constexpr (Q < NQ) {
    float c, s;
    __sincosf(f[Q] * 0.5f, &s, &c);
    float2 d  = make_float2(c, 0.0f);
    float2 od = make_float2(0.0f, -s);         // -i sin(a/2)
    apply_1q<9 - Q>(st, lane, d, od, od, d);
    embed_rx<Q + 1>(st, f, lane);
  }
}

template <int Q>
__device__ __forceinline__ void layer_rots(float2 (&st)[32],
                                           const float* __restrict__ qwl, int lane) {
  if constexpr (Q < NQ) {
    float2 m00, m01, m10, m11;
    rot_mat(qwl[Q * 3 + 0], qwl[Q * 3 + 1], qwl[Q * 3 + 2], m00, m01, m10, m11);
    apply_1q<9 - Q>(st, lane, m00, m01, m10, m11);
    layer_rots<Q + 1>(st, qwl, lane);
  }
}

template <int R, int Q>
__device__ __forceinline__ void layer_cnots(float2 (&st)[32], int lane) {
  if constexpr (Q < NQ) {
    constexpr int T = (Q + R) % NQ;
    apply_cnot<9 - Q, 9 - T>(st, lane);
    layer_cnots<R, Q + 1>(st, lane);
  }
}

__global__ __launch_bounds__(256) void qsim_fused(
    const float* __restrict__ feats,  // (8192, 16); cols 0..9 used
    const float* __restrict__ qw,     // (6, 10, 3)
    const float* __restrict__ W2,     // (64, 10)
    const float* __restrict__ b2,     // (64)
    float* __restrict__ out)          // (8192, 64)
{
  const int lane = threadIdx.x & 31;
  const int samp = (blockIdx.x * blockDim.x + threadIdx.x) >> 5;  // one wave/sample

  // |0...0>
  float2 st[32];
  #pragma unroll
  for (int r = 0; r < 32; ++r) st[r] = make_float2(0.0f, 0.0f);
  st[0].x = (lane == 0) ? 1.0f : 0.0f;

  // AngleEmbedding (RX per qubit)
  embed_rx<0>(st, feats + (size_t)samp * FEAT_LD, lane);

  // StronglyEntanglingLayers: Rot x10 then ring of CNOTs (range r = l+1 for l<6)
  for (int l = 0; l < NL; ++l) {
    layer_rots<0>(st, qw + l * NQ * 3, lane);
    switch (l) {
      case 0: layer_cnots<1, 0>(st, lane); break;
      case 1: layer_cnots<2, 0>(st, lane); break;
      case 2: layer_cnots<3, 0>(st, lane); break;
      case 3: layer_cnots<4, 0>(st, lane); break;
      case 4: layer_cnots<5, 0>(st, lane); break;
      default: layer_cnots<6, 0>(st, lane); break;
    }
  }

  // <Z_q> = sum_i sign(bit p of i) * |amp_i|^2
  float z[NQ];
  #pragma unroll
  for (int q = 0; q < NQ; ++q) z[q] = 0.0f;
  float tot = 0.0f;
  #pragma unroll
  for (int r = 0; r < 32; ++r) {
    float p = st[r].x * st[r].x + st[r].y * st[r].y;
    tot += p;
    #pragma unroll
    for (int q = 5; q < NQ; ++q)            // register-bit qubits
      z[q] += ((r >> (9 - q)) & 1) ? -p : p;
  }
  #pragma unroll
  for (int q = 0; q < 5; ++q)               // lane-bit qubits
    z[q] = ((lane >> (4 - q)) & 1) ? -tot : tot;

  // wave-wide butterfly reduction (all lanes end with full z[q])
  #pragma unroll
  for (int q = 0; q < NQ; ++q) {
    #pragma unroll
    for (int m = 16; m >= 1; m >>= 1) z[q] += __shfl_xor(z[q], m, 32);
  }

  // fused epilogue: out[samp, :] = z @ W2.T + b2 ; each lane writes 2 outputs
  #pragma unroll
  for (int half = 0; half < 2; ++half) {
    const int j = lane + half * 32;
    float o = b2[j];
    #pragma unroll
    for (int q = 0; q < NQ; ++q) o = fmaf(z[q], W2[j * NQ + q], o);
    out[(size_t)samp * DOUT + j] = o;
  }
}

// ===========================================================================
extern "C" void kernel_launch(void* const* d_in, const int* in_sizes, int n_in,
                              void* d_out, int out_size, void* d_ws, size_t ws_size,
                              hipStream_t stream) {
  (void)in_sizes; (void)n_in; (void)out_size; (void)ws_size;
  const float* x  = (const float*)d_in[0];  // (8192, 256)
  const float* W1 = (const float*)d_in[1];  // (128, 256)
  const float* b1 = (const float*)d_in[2];  // (128)
  const float* qw = (const float*)d_in[3];  // (6, 10, 3)
  const float* W2 = (const float*)d_in[4];  // (64, 10)
  const float* b2 = (const float*)d_in[5];  // (64)
  float* out   = (float*)d_out;             // (8192, 64)
  float* feats = (float*)d_ws;              // (8192, 16) = 512 KB scratch

  // 512 sample-tiles of 16, 8 waves/block -> 64 blocks
  gemm1_relu_wmma<<<dim3(64), dim3(256), 0, stream>>>(x, W1, b1, feats);
  // one wave per sample, 8 waves/block -> 1024 blocks
  qsim_fused<<<dim3(1024), dim3(256), 0, stream>>>(feats, qw, W2, b2, out);
}